// HierarchicalModel_56788057588165
// MI455X (gfx1250) — compile-verified
//
#include <hip/hip_runtime.h>
#include <hip/hip_bf16.h>

typedef __attribute__((ext_vector_type(2))) float v2f;
typedef __attribute__((ext_vector_type(8))) float v8f;

#define S_  128
#define T_  48
#define HS_ 128
#define V_  32000
#define NROW (47 * 128)          // 6016 decoder hidden rows
#define NTILE_V (V_ / 16)        // 2000 vocab tiles

__device__ __forceinline__ v8f wmma4(v2f a, v2f b, v8f c) {
  // D = A(16x4 f32) * B(4x16 f32) + C(16x16 f32)
  return __builtin_amdgcn_wmma_f32_16x16x4_f32(
      false, a, false, b, (short)0, c, false, false);
}

__device__ __forceinline__ float sigmoidf_(float x) {
  return 1.0f / (1.0f + __expf(-x));
}

// ---------------------------------------------------------------------------
// zero helper
// ---------------------------------------------------------------------------
__global__ void zero_kernel(float* __restrict__ p, int n) {
  int i = blockIdx.x * blockDim.x + threadIdx.x;
  if (i < n) p[i] = 0.0f;
}

// ---------------------------------------------------------------------------
// One GRU step for batch 128, hidden 128, WMMA f32 16x16x4.
// Block = 1 wave (32 lanes), grid = (8 batch tiles, 8 hidden tiles).
// gi comes from an embedding gather (K=64); decoder's context half of gi is a
// precomputed per-gate bias `extra`. gh comes from h_in (K=128).
// ---------------------------------------------------------------------------
__global__ __launch_bounds__(32) void gru_step_wmma(
    const int* __restrict__ doc, int t, int T,
    const float* __restrict__ emb,        // (V, 64)
    const float* __restrict__ Wih, int ldWih,  // (384, ldWih), emb part first 64
    const float* __restrict__ Whh,        // (384, 128)
    const float* __restrict__ bih, const float* __restrict__ bhh,
    const float* __restrict__ extra,      // (384,) or nullptr
    const float* __restrict__ h_in,       // (128,128)
    float* __restrict__ h_out)            // (128,128)
{
  const int lane = threadIdx.x;
  const int n = lane & 15;          // also the A-operand row M for this lane
  const int half = lane >> 4;
  const int s0 = blockIdx.x * 16;
  const int j0 = blockIdx.y * 16;

  const int tok = doc[(s0 + n) * T + t];
  const float* embrow = emb + (long)tok * 64;

  v8f accR = {}, accZ = {}, accIN = {}, accHN = {};

  // gi = x @ Wih[:, :64].T   (K = 64)
  {
    const float* WR = Wih + (long)(      j0 + n) * ldWih;
    const float* WZ = Wih + (long)(128 + j0 + n) * ldWih;
    const float* WN = Wih + (long)(256 + j0 + n) * ldWih;
#pragma unroll
    for (int kk = 0; kk < 16; ++kk) {
      const int k = kk * 4 + half * 2;
      v2f a; a.x = embrow[k]; a.y = embrow[k + 1];
      v2f b;
      b.x = WR[k]; b.y = WR[k + 1]; accR  = wmma4(a, b, accR);
      b.x = WZ[k]; b.y = WZ[k + 1]; accZ  = wmma4(a, b, accZ);
      b.x = WN[k]; b.y = WN[k + 1]; accIN = wmma4(a, b, accIN);
    }
  }
  // gh = h @ Whh.T           (K = 128)
  {
    const float* hrow = h_in + (long)(s0 + n) * HS_;
    const float* WR = Whh + (long)(      j0 + n) * HS_;
    const float* WZ = Whh + (long)(128 + j0 + n) * HS_;
    const float* WN = Whh + (long)(256 + j0 + n) * HS_;
#pragma unroll
    for (int kk = 0; kk < 32; ++kk) {
      const int k = kk * 4 + half * 2;
      v2f a; a.x = hrow[k]; a.y = hrow[k + 1];
      v2f b;
      b.x = WR[k]; b.y = WR[k + 1]; accR  = wmma4(a, b, accR);
      b.x = WZ[k]; b.y = WZ[k + 1]; accZ  = wmma4(a, b, accZ);
      b.x = WN[k]; b.y = WN[k + 1]; accHN = wmma4(a, b, accHN);
    }
  }

  float bR = bih[j0 + n] + bhh[j0 + n];
  float bZ = bih[128 + j0 + n] + bhh[128 + j0 + n];
  float bI = bih[256 + j0 + n];
  float bH = bhh[256 + j0 + n];
  if (extra) {
    bR += extra[j0 + n];
    bZ += extra[128 + j0 + n];
    bI += extra[256 + j0 + n];
  }

#pragma unroll
  for (int v = 0; v < 8; ++v) {
    const int m = v + half * 8;          // C/D layout row
    float r  = sigmoidf_(accR[v] + bR);
    float z  = sigmoidf_(accZ[v] + bZ);
    float nn = tanhf(accIN[v] + bI + r * (accHN[v] + bH));
    float ho = h_in[(long)(s0 + m) * HS_ + j0 + n];
    h_out[(long)(s0 + m) * HS_ + j0 + n] = (1.0f - z) * nn + z * ho;
  }
}

// ---------------------------------------------------------------------------
// Document-level GRU, batch 1, 128 steps. One block of 384 threads (VALU).
// ---------------------------------------------------------------------------
__global__ __launch_bounds__(384) void doc_gru_kernel(
    const float* __restrict__ sent,      // (128,128)
    const float* __restrict__ Wih, const float* __restrict__ Whh, // (384,128)
    const float* __restrict__ bih, const float* __restrict__ bhh,
    float* __restrict__ out)             // (128,)
{
  __shared__ float h[HS_];
  __shared__ float gi[3 * HS_];
  __shared__ float gh[3 * HS_];
  const int g = threadIdx.x;
  if (g < HS_) h[g] = 0.0f;
  __syncthreads();

  const float* wi = Wih + (long)g * HS_;
  const float* wh = Whh + (long)g * HS_;
  for (int s = 0; s < S_; ++s) {
    const float* sv = sent + (long)s * HS_;
    float a = 0.0f, c = 0.0f;
#pragma unroll 8
    for (int k = 0; k < HS_; ++k) { a += sv[k] * wi[k]; c += h[k] * wh[k]; }
    gi[g] = a + bih[g];
    gh[g] = c + bhh[g];
    __syncthreads();
    if (g < HS_) {
      float r  = sigmoidf_(gi[g] + gh[g]);
      float z  = sigmoidf_(gi[128 + g] + gh[128 + g]);
      float nn = tanhf(gi[256 + g] + r * gh[256 + g]);
      h[g] = (1.0f - z) * nn + z * h[g];
    }
    __syncthreads();
  }
  if (g < HS_) out[g] = h[g];
}

// ---------------------------------------------------------------------------
// ctx_gi[g] = sum_k doc_vec[k] * dec_Wih[g, 64+k]   (constant over batch/time)
// ---------------------------------------------------------------------------
__global__ void ctx_kernel(const float* __restrict__ doc_vec,
                           const float* __restrict__ dec_Wih,  // (384,192)
                           float* __restrict__ ctx) {
  int g = blockIdx.x * blockDim.x + threadIdx.x;
  if (g < 384) {
    const float* w = dec_Wih + (long)g * 192 + 64;
    float s = 0.0f;
#pragma unroll 8
    for (int k = 0; k < HS_; ++k) s += doc_vec[k] * w[k];
    ctx[g] = s;
  }
}

// ---------------------------------------------------------------------------
// Fused vocab projection + log-softmax NLL.
// Block owns 32 rows of H (two 16-row A register sets that SHARE every B
// load, doubling FLOP per byte of L2 traffic on fc_W). Each of the 8 waves
// sweeps the 2000 vocab tiles with WMMA. Logits are bounded (|h|<1,
// |w|~0.05), so sum(exp(val)) without max-shifting is exact in fp32.
// ---------------------------------------------------------------------------
__global__ __launch_bounds__(256) void fc_loss_kernel(
    const float* __restrict__ H,         // (NROW, 128)
    const int* __restrict__ doc, int T,
    const float* __restrict__ W,         // (V, 128)
    const float* __restrict__ b,         // (V,)
    float* __restrict__ accum)
{
  const int lane = threadIdx.x & 31;
  const int wave = threadIdx.x >> 5;
  const int n = lane & 15;
  const int half = lane >> 4;
  const int r0 = blockIdx.x * 32;

  // Two A operand sets resident in registers: A[set][m][k] = H[r0+set*16+m][k]
  float areg[2][64];
#pragma unroll
  for (int set = 0; set < 2; ++set) {
    const float* Arow = H + (long)(r0 + set * 16 + n) * HS_;
#pragma unroll
    for (int kk = 0; kk < 32; ++kk) {
      const int k = kk * 4 + half * 2;
      areg[set][2 * kk]     = Arow[k];
      areg[set][2 * kk + 1] = Arow[k + 1];
    }
  }
  // targets for the 16 rows this lane owns in the two C/D layouts
  int tgtIdx[2][8];
#pragma unroll
  for (int set = 0; set < 2; ++set)
#pragma unroll
    for (int v = 0; v < 8; ++v) {
      const int row = r0 + set * 16 + v + half * 8;
      const int tt = row >> 7, ss = row & 127;
      tgtIdx[set][v] = doc[ss * T + tt + 1];
    }

  float rsum[2][8], tgt[2][8];
#pragma unroll
  for (int set = 0; set < 2; ++set)
#pragma unroll
    for (int v = 0; v < 8; ++v) { rsum[set][v] = 0.0f; tgt[set][v] = 0.0f; }

  for (int tile = wave; tile < NTILE_V; tile += 8) {
    const int c0 = tile * 16;
    const float* Wrow = W + (long)(c0 + n) * HS_;
    v8f acc0 = {}, acc1 = {};
#pragma unroll
    for (int kk = 0; kk < 32; ++kk) {
      const int k = kk * 4 + half * 2;
      v2f bb; bb.x = Wrow[k]; bb.y = Wrow[k + 1];
      v2f a0; a0.x = areg[0][2 * kk]; a0.y = areg[0][2 * kk + 1];
      v2f a1; a1.x = areg[1][2 * kk]; a1.y = areg[1][2 * kk + 1];
      acc0 = wmma4(a0, bb, acc0);      // independent accumulators back-to-back
      acc1 = wmma4(a1, bb, acc1);      // also cover WMMA->WMMA hazard slots
    }
    const float bias = b[c0 + n];
#pragma unroll
    for (int v = 0; v < 8; ++v) {
      const float v0 = acc0[v] + bias;
      const float v1 = acc1[v] + bias;
      rsum[0][v] += __expf(v0);
      rsum[1][v] += __expf(v1);
      tgt[0][v] += (c0 + n == tgtIdx[0][v]) ? v0 : 0.0f;
      tgt[1][v] += (c0 + n == tgtIdx[1][v]) ? v1 : 0.0f;
    }
  }

  __shared__ float ssum[256][16];
  __shared__ float stgt[256][16];
#pragma unroll
  for (int set = 0; set < 2; ++set)
#pragma unroll
    for (int v = 0; v < 8; ++v) {
      ssum[threadIdx.x][set * 8 + v] = rsum[set][v];
      stgt[threadIdx.x][set * 8 + v] = tgt[set][v];
    }
  __syncthreads();

  __shared__ float rowloss[32];
  if (threadIdx.x < 32) {
    const int m2 = threadIdx.x;          // row r0+m2
    const int set = m2 >> 4;
    const int mloc = m2 & 15;
    const int vv = mloc & 7, hh = mloc >> 3;
    const int slot = set * 8 + vv;
    float Sx = 0.0f, Tg = 0.0f;
    for (int w = 0; w < 8; ++w) {
      for (int j = 0; j < 16; ++j) {
        const int idx = w * 32 + hh * 16 + j;
        Sx += ssum[idx][slot];
        Tg += stgt[idx][slot];
      }
    }
    rowloss[m2] = __logf(Sx) - Tg;       // -log p[target] for this row
  }
  __syncthreads();
  if (threadIdx.x == 0) {
    float s = 0.0f;
    for (int m = 0; m < 32; ++m) s += rowloss[m];
    atomicAdd(accum, s);
  }
}

__global__ void finalize_kernel(const float* __restrict__ accum,
                                float* __restrict__ out) {
  if (threadIdx.x == 0) out[0] = accum[0] * (1.0f / (float)NROW);
}

// ---------------------------------------------------------------------------
extern "C" void kernel_launch(void* const* d_in, const int* in_sizes, int n_in,
                              void* d_out, int out_size, void* d_ws, size_t ws_size,
                              hipStream_t stream) {
  const int*   doc      = (const int*)  d_in[0];
  const float* enc_emb  = (const float*)d_in[1];
  const float* enc_Wih  = (const float*)d_in[2];
  const float* enc_Whh  = (const float*)d_in[3];
  const float* enc_bih  = (const float*)d_in[4];
  const float* enc_bhh  = (const float*)d_in[5];
  const float* doc_Wih  = (const float*)d_in[6];
  const float* doc_Whh  = (const float*)d_in[7];
  const float* doc_bih  = (const float*)d_in[8];
  const float* doc_bhh  = (const float*)d_in[9];
  const float* dec_emb  = (const float*)d_in[10];
  const float* dec_Wih  = (const float*)d_in[11];
  const float* dec_Whh  = (const float*)d_in[12];
  const float* dec_bih  = (const float*)d_in[13];
  const float* dec_bhh  = (const float*)d_in[14];
  const float* fc_W     = (const float*)d_in[15];
  const float* fc_b     = (const float*)d_in[16];

  float* ws      = (float*)d_ws;
  float* hA      = ws;                  // 128*128
  float* hB      = ws + 16384;          // 128*128
  float* doc_vec = ws + 32768;          // 128
  float* ctx     = ws + 32896;          // 384
  float* accum   = ws + 33280;          // 1
  float* Hst     = ws + 33296;          // 47*128*128

  // per-call init (graph replays don't re-zero anything)
  zero_kernel<<<(16384 + 255) / 256, 256, 0, stream>>>(hA, 16384);
  zero_kernel<<<1, 32, 0, stream>>>(accum, 1);

  // encoder scan: 48 sequential steps, ping-pong hA/hB (ends in hA)
  for (int t = 0; t < T_; ++t) {
    const float* hin  = (t & 1) ? hB : hA;
    float*       hout = (t & 1) ? hA : hB;
    gru_step_wmma<<<dim3(8, 8), 32, 0, stream>>>(
        doc, t, T_, enc_emb, enc_Wih, 64, enc_Whh, enc_bih, enc_bhh,
        nullptr, hin, hout);
  }
  const float* sent_vec = hA;

  doc_gru_kernel<<<1, 384, 0, stream>>>(sent_vec, doc_Wih, doc_Whh,
                                        doc_bih, doc_bhh, doc_vec);
  ctx_kernel<<<2, 192, 0, stream>>>(doc_vec, dec_Wih, ctx);

  // decoder scan: 47 steps, states written straight into Hst
  for (int t = 0; t < T_ - 1; ++t) {
    const float* hin = (t == 0) ? sent_vec : (Hst + (long)(t - 1) * 16384);
    gru_step_wmma<<<dim3(8, 8), 32, 0, stream>>>(
        doc, t, T_, dec_emb, dec_Wih, 192, dec_Whh, dec_bih, dec_bhh,
        ctx, hin, Hst + (long)t * 16384);
  }

  fc_loss_kernel<<<NROW / 32, 256, 0, stream>>>(Hst, doc, T_, fc_W, fc_b, accum);
  finalize_kernel<<<1, 32, 0, stream>>>(accum, (float*)d_out);
}